// QL_MATCC_GNN_Model_25056839205730
// MI455X (gfx1250) — compile-verified
//
#include <hip/hip_runtime.h>
#include <math.h>

typedef __attribute__((ext_vector_type(2))) float v2f;
typedef __attribute__((ext_vector_type(8))) float v8f;

#define N_NODES 1024
#define F_DIM   64
#define GAT_ALPHA 0.2f
#define NEG_INF_F (-9.0e15f)
#define PS 1028   // padded LDS row stride (floats): 1028 % 64 = 4 -> conflict-free column reads

// ---------------------------------------------------------------------------
// Kernel 1: Wh = h @ W via f32 WMMA, with e1 = Wh@a1 and e2 = Wh@a2 fused
// from the accumulator registers (no second pass over Wh, no extra launch).
// One wave per 16-row tile; 4 column tiles (NOT unrolled -> one live
// accumulator, no spills) x 16 K-steps of v_wmma_f32_16x16x4_f32.
// D layout: VGPR r -> (M = r + 8*half, N = lane&15); a 16-lane xor-tree
// (masks 1,2,4,8 stay inside each half) reduces across N for fixed M.
// ---------------------------------------------------------------------------
__global__ void k_wh_e(const float* __restrict__ h, const float* __restrict__ W,
                       const float* __restrict__ a,
                       float* __restrict__ Wh, float* __restrict__ e1,
                       float* __restrict__ e2) {
  const int lane = threadIdx.x & 31;
  const int half = lane >> 4;
  const int l    = lane & 15;
  const int m0   = blockIdx.x * 16;

  float p1[8] = {0.f, 0.f, 0.f, 0.f, 0.f, 0.f, 0.f, 0.f};
  float p2[8] = {0.f, 0.f, 0.f, 0.f, 0.f, 0.f, 0.f, 0.f};

#pragma unroll 1            // keep ONE accumulator live -> no scratch spills
  for (int nt = 0; nt < 4; ++nt) {
    const int ncol = nt * 16 + l;
    v8f c = {};
#pragma unroll
    for (int kb = 0; kb < F_DIM; kb += 4) {
      const int k0 = kb + 2 * half;
      v2f av = *(const v2f*)(h + (m0 + l) * F_DIM + k0);   // consecutive K pair
      v2f bv;
      bv.x = W[k0 * F_DIM + ncol];
      bv.y = W[(k0 + 1) * F_DIM + ncol];
      c = __builtin_amdgcn_wmma_f32_16x16x4_f32(false, av, false, bv,
                                                (short)0, c, false, false);
    }
    const float a1n = a[ncol];
    const float a2n = a[F_DIM + ncol];
#pragma unroll
    for (int r = 0; r < 8; ++r) {
      Wh[(m0 + r + 8 * half) * F_DIM + ncol] = c[r];
      p1[r] += c[r] * a1n;
      p2[r] += c[r] * a2n;
    }
  }

  // Reduce e-partials across the 16 columns held by each half-wave.
#pragma unroll
  for (int off = 1; off < 16; off <<= 1) {
#pragma unroll
    for (int r = 0; r < 8; ++r) {
      p1[r] += __shfl_xor(p1[r], off, 32);
      p2[r] += __shfl_xor(p2[r], off, 32);
    }
  }
  if (l == 0) {
#pragma unroll
    for (int r = 0; r < 8; ++r) {
      const int m = m0 + r + 8 * half;
      e1[m] = p1[r];
      e2[m] = p2[r];
    }
  }
}

// ---------------------------------------------------------------------------
// Kernel 2: fused masked-softmax + attention@Wh + ELU.
// 64 blocks x 128 threads (4 wave32s); block handles 16 output rows.
// Phase A: vectorized (b128) adj/e2 streams -> masked leaky-relu scores in
//          LDS, wave-reduced max, exp(s-max) written back, row sums kept.
// Phase B: each wave owns one 16-col N-tile; 256 x v_wmma_f32_16x16x4_f32
//          with A-frags as ds_load_b64 from LDS, B (=Wh) from global (L2).
// ---------------------------------------------------------------------------
__global__ void k_attn(const float* __restrict__ Wh, const int* __restrict__ adj,
                       const float* __restrict__ e1, const float* __restrict__ e2,
                       float* __restrict__ out) {
  extern __shared__ float smem[];        // 16*PS attention tile + 16 row sums
  float* P    = smem;
  float* sums = smem + 16 * PS;

  const int tid  = threadIdx.x;
  const int wave = tid >> 5;
  const int lane = tid & 31;
  const int half = lane >> 4;
  const int l    = lane & 15;
  const int i0   = blockIdx.x * 16;

  // ---- Phase A: masked scores -> softmax numerators in LDS ----
#pragma unroll 1
  for (int rr = 0; rr < 4; ++rr) {
    const int r = wave * 4 + rr;
    const int i = i0 + r;
    const float ei = e1[i];

    float mx = NEG_INF_F;
#pragma unroll
    for (int t = 0; t < N_NODES / 128; ++t) {        // 8 iters, 16B/lane each
      const int j0 = (t * 32 + lane) * 4;
      const int4   av = *(const int4*)(adj + (size_t)i * N_NODES + j0);
      const float4 ev = *(const float4*)(e2 + j0);
      float4 s;
      s.x = ei + ev.x; s.y = ei + ev.y; s.z = ei + ev.z; s.w = ei + ev.w;
      s.x = (s.x > 0.f) ? s.x : GAT_ALPHA * s.x;
      s.y = (s.y > 0.f) ? s.y : GAT_ALPHA * s.y;
      s.z = (s.z > 0.f) ? s.z : GAT_ALPHA * s.z;
      s.w = (s.w > 0.f) ? s.w : GAT_ALPHA * s.w;
      s.x = (av.x > 0) ? s.x : NEG_INF_F;
      s.y = (av.y > 0) ? s.y : NEG_INF_F;
      s.z = (av.z > 0) ? s.z : NEG_INF_F;
      s.w = (av.w > 0) ? s.w : NEG_INF_F;
      *(float4*)(P + r * PS + j0) = s;               // 16B-aligned (PS%4==0)
      mx = fmaxf(mx, fmaxf(fmaxf(s.x, s.y), fmaxf(s.z, s.w)));
      if (i + 1 < N_NODES)                           // stream next adj row ahead
        __builtin_prefetch(adj + (size_t)(i + 1) * N_NODES + j0, 0, 1);
    }
#pragma unroll
    for (int off = 16; off > 0; off >>= 1)
      mx = fmaxf(mx, __shfl_xor(mx, off, 32));

    float sum = 0.0f;
#pragma unroll
    for (int t = 0; t < N_NODES / 128; ++t) {
      const int j0 = (t * 32 + lane) * 4;
      float4 p = *(const float4*)(P + r * PS + j0);
      p.x = __expf(p.x - mx);                        // masked entries -> 0
      p.y = __expf(p.y - mx);
      p.z = __expf(p.z - mx);
      p.w = __expf(p.w - mx);
      *(float4*)(P + r * PS + j0) = p;
      sum += p.x + p.y + p.z + p.w;
    }
#pragma unroll
    for (int off = 16; off > 0; off >>= 1)
      sum += __shfl_xor(sum, off, 32);
    if (lane == 0) sums[r] = sum;
  }
  __syncthreads();

  // ---- Phase B: (attention tile) @ Wh via f32 WMMA ----
  const int ncol = wave * 16 + l;
  v8f c = {};
#pragma unroll 4
  for (int kb = 0; kb < N_NODES; kb += 4) {
    const int k0 = kb + 2 * half;
    const v2f av = *(const v2f*)(P + l * PS + k0);   // ds_load_b64 (8B aligned)
    v2f bv;
    bv.x = Wh[k0 * F_DIM + ncol];
    bv.y = Wh[(k0 + 1) * F_DIM + ncol];
    c = __builtin_amdgcn_wmma_f32_16x16x4_f32(false, av, false, bv,
                                              (short)0, c, false, false);
  }

#pragma unroll
  for (int r = 0; r < 8; ++r) {
    const int m = r + 8 * half;
    float v = c[r] / sums[m];                        // softmax normalization
    v = (v > 0.0f) ? v : (__expf(v) - 1.0f);         // ELU (alpha=1)
    out[(i0 + m) * F_DIM + ncol] = v;
  }
}

// ---------------------------------------------------------------------------
extern "C" void kernel_launch(void* const* d_in, const int* in_sizes, int n_in,
                              void* d_out, int out_size, void* d_ws, size_t ws_size,
                              hipStream_t stream) {
  const float* h   = (const float*)d_in[0];   // (1024, 64) f32
  const int*   adj = (const int*)  d_in[1];   // (1024, 1024) i32
  const float* W   = (const float*)d_in[2];   // (64, 64) f32
  const float* a   = (const float*)d_in[3];   // (128, 1) f32
  float* out = (float*)d_out;                 // (1024, 64) f32

  float* Wh = (float*)d_ws;                   // 1024*64 f32 = 256 KB
  float* e1 = Wh + N_NODES * F_DIM;           // 1024 f32
  float* e2 = e1 + N_NODES;                   // 1024 f32

  k_wh_e<<<N_NODES / 16, 32, 0, stream>>>(h, W, a, Wh, e1, e2);

  const size_t shmem = (size_t)(16 * PS + 16) * sizeof(float);
  k_attn<<<N_NODES / 16, 128, shmem, stream>>>(Wh, adj, e1, e2, out);
}